// LabelCls_38534446579952
// MI455X (gfx1250) — compile-verified
//
#include <hip/hip_runtime.h>

// ---------------- problem constants (match reference) ----------------
#define BATCH   64
#define NCASC   3
#define NROI    65536
#define ROWS    (BATCH * NCASC)          // 192
#define NELEM   (ROWS * NROI)            // 12,582,912
#define TPB     256                      // 8 wave32 per block
#define TILE    2048                     // floats staged per tile (8 KB)
#define NTILES  (NROI / TILE)            // 32
#define EPT     (TILE / TPB)             // 8 elements / thread / tile
#define V4PT    (TILE / (TPB * 4))       // 2 b128 async issues / thread / tile
#define K_POS   16
#define K_NEG   48
#define NEG_T   0.3f

// ---------------- helpers ----------------

// deterministic counter-based uniform u32 (stands in for jax threefry stream)
__device__ __forceinline__ unsigned rand_u32(unsigned idx) {
  unsigned x = idx + 0x9E3779B9u;
  x ^= x >> 16; x *= 0x7feb352du;
  x ^= x >> 15; x *= 0x846ca68bu;
  x ^= x >> 16;
  return x;
}

// monotone float <-> u32 key (order-preserving, handles sign bit)
__device__ __forceinline__ unsigned f2key(float f) {
  unsigned b = __float_as_uint(f);
  return b ^ ((b >> 31) ? 0xFFFFFFFFu : 0x80000000u);
}
__device__ __forceinline__ float key2f(unsigned k) {
  unsigned b = (k >> 31) ? (k ^ 0x80000000u) : ~k;
  return __uint_as_float(b);
}

// CDNA5 async global->LDS staging of one TILE of floats (b128 per lane).
// Each wave issues V4PT instructions per array -> ASYNCcnt += V4PT.
__device__ __forceinline__ void stage_tile_async(const float* __restrict__ g,
                                                 float* s) {
  const int t = threadIdx.x;
#pragma unroll
  for (int j = 0; j < V4PT; ++j) {
    const int i4 = t + j * TPB;
    unsigned lds_off = (unsigned)(unsigned long long)(s + i4 * 4); // low 32 = LDS byte addr
    const float* gp = g + (size_t)i4 * 4;
    asm volatile("global_load_async_to_lds_b128 %0, %1, off"
                 : : "v"(lds_off), "v"(gp) : "memory");
  }
}
// Wait until at most N async ops outstanding (per-wave, in-order completion):
// with N = #ops just issued for the NEXT tile, the CURRENT tile is complete.
template <int N>
__device__ __forceinline__ void async_wait_le() {
  asm volatile("s_wait_asynccnt %0" : : "i"(N) : "memory");
}

// ---------------- exact k-th largest via 4x8-bit radix select ----------------
// mode 0 (POS): key over ALL 65536 masked values v_i (zeros included) ->
//               exactly jax.lax.top_k(iou_masked,k)[k-1].
// mode 1 (NEG): key = rand among eligible (overlap <= 0.3); if fewer than k
//               eligible, returns 0 (i.e. "-inf kth": select all eligible).
// Double-buffered async pipeline: tile t+1 streams into LDS while tile t is
// histogrammed.
__device__ unsigned radix_select(int mode, unsigned k,
                                 const float* __restrict__ ovR,
                                 const float* __restrict__ ioR,
                                 unsigned rowbase, unsigned posc, float rmax,
                                 float posT,
                                 float* s_ov, float* s_io,   // [2*TILE] each
                                 unsigned* s_hist, unsigned* s_bc) {
  unsigned prefix = 0;
  int pbits = 0;
  unsigned want = k;
  for (int pass = 0; pass < 4; ++pass) {
    s_hist[threadIdx.x & 255] = 0;
    __syncthreads();
    const int shift = 24 - 8 * pass;
    // prologue: stage tile 0 into buffer 0
    stage_tile_async(ovR, s_ov);
    if (mode == 0) stage_tile_async(ioR, s_io);
    for (int tile = 0; tile < NTILES; ++tile) {
      const int cur = tile & 1;
      if (tile + 1 < NTILES) {
        stage_tile_async(ovR + (tile + 1) * TILE, s_ov + (cur ^ 1) * TILE);
        if (mode == 0) {
          stage_tile_async(ioR + (tile + 1) * TILE, s_io + (cur ^ 1) * TILE);
          async_wait_le<2 * V4PT>();     // tile 'tile' fully landed
        } else {
          async_wait_le<V4PT>();
        }
      } else {
        async_wait_le<0>();
      }
      __syncthreads();                   // all waves' portions of tile visible
      const float* pov = s_ov + cur * TILE;
      const float* pio = s_io + cur * TILE;
#pragma unroll
      for (int kk = 0; kk < EPT; ++kk) {
        const int li = threadIdx.x + (kk << 8);
        unsigned key;
        bool m;
        if (mode == 0) {
          const float o = pov[li], x = pio[li];
          const float v = (posc > 0u) ? ((o >= posT) ? x : 0.0f)
                                      : ((x >= rmax) ? x : 0.0f);
          key = f2key(v);
          m = true;
        } else {
          m = (pov[li] <= NEG_T);
          key = rand_u32(rowbase + (unsigned)(tile * TILE + li));
        }
        if (m && (pass == 0 || (key >> (32 - pbits)) == prefix))
          atomicAdd(&s_hist[(key >> shift) & 255], 1u);
      }
      __syncthreads();                   // readers done before buffer reuse
    }
    if (threadIdx.x == 0) {
      unsigned cum = 0;
      int sel = -1;
      for (int b = 255; b >= 0; --b) {
        const unsigned c = s_hist[b];
        if (cum + c >= want) { sel = b; break; }
        cum += c;
      }
      if (sel < 0) { s_bc[0] = 1u; s_bc[1] = 0u; s_bc[2] = 0u; }
      else {
        s_bc[0] = 0u;
        s_bc[1] = (prefix << 8) | (unsigned)sel;
        s_bc[2] = want - cum;            // rank within selected bin
      }
    }
    __syncthreads();
    if (s_bc[0]) return 0u;              // fewer than k candidates (NEG only)
    prefix = s_bc[1];
    want = s_bc[2];
    pbits += 8;
    __syncthreads();
  }
  return prefix;                          // exact 32-bit k-th largest key
}

// ---------------- kernel 1: per-row statistics + thresholds ----------------
__global__ __launch_bounds__(TPB) void select_kernel(
    const float* __restrict__ ov, const float* __restrict__ io,
    float* __restrict__ thrA, float* __restrict__ top2A,
    float* __restrict__ rmaxA, unsigned* __restrict__ poscA,
    unsigned* __restrict__ kthnA) {
  __shared__ float s_ov[2 * TILE];
  __shared__ float s_io[2 * TILE];
  __shared__ unsigned s_hist[256];
  __shared__ float s_redf[TPB];
  __shared__ unsigned s_redu[TPB];
  __shared__ unsigned s_bc[4];

  const int r = blockIdx.x;              // row = b*3 + h
  const int h = r % 3;
  const float posT = (h == 0) ? 0.6f : ((h == 1) ? 0.7f : 0.8f);
  const float iouT = (h == 0) ? 0.2f : ((h == 1) ? 0.3f : 0.4f);
  const float* ovR = ov + (size_t)r * NROI;
  const float* ioR = io + (size_t)r * NROI;
  const unsigned rowbase = (unsigned)r * (unsigned)NROI;
  const int tid = threadIdx.x;

  // ---- phase 0: pos_count (overlap >= posT) and row max(iou), pipelined ----
  float lmax = -__builtin_inff();
  unsigned lpos = 0;
  stage_tile_async(ovR, s_ov);
  stage_tile_async(ioR, s_io);
  for (int tile = 0; tile < NTILES; ++tile) {
    const int cur = tile & 1;
    if (tile + 1 < NTILES) {
      stage_tile_async(ovR + (tile + 1) * TILE, s_ov + (cur ^ 1) * TILE);
      stage_tile_async(ioR + (tile + 1) * TILE, s_io + (cur ^ 1) * TILE);
      async_wait_le<2 * V4PT>();
    } else {
      async_wait_le<0>();
    }
    __syncthreads();
    const float* pov = s_ov + cur * TILE;
    const float* pio = s_io + cur * TILE;
#pragma unroll
    for (int kk = 0; kk < EPT; ++kk) {
      const int li = tid + (kk << 8);
      lpos += (pov[li] >= posT) ? 1u : 0u;
      lmax = fmaxf(lmax, pio[li]);
    }
    __syncthreads();
  }
  s_redf[tid] = lmax;
  s_redu[tid] = lpos;
  __syncthreads();
  for (int s = TPB / 2; s > 0; s >>= 1) {
    if (tid < s) {
      s_redf[tid] = fmaxf(s_redf[tid], s_redf[tid + s]);
      s_redu[tid] += s_redu[tid + s];
    }
    __syncthreads();
  }
  const float rmax = s_redf[0];
  const unsigned posc = s_redu[0];
  __syncthreads();

  // ---- phase 1: 16th largest of iou_masked -> positive threshold ----
  const unsigned kp16 = radix_select(0, K_POS, ovR, ioR, rowbase, posc, rmax,
                                     posT, s_ov, s_io, s_hist, s_bc);
  const float kth16 = key2f(kp16);
  const float thrv = (kth16 >= iouT) ? kth16 : iouT;

  // ---- phase 2 (h==0 only): 2nd largest (MAX_POS) ----
  float t2 = __builtin_inff();
  if (h == 0) {
    const unsigned kp2 = radix_select(0, 2u, ovR, ioR, rowbase, posc, rmax,
                                      posT, s_ov, s_io, s_hist, s_bc);
    t2 = key2f(kp2);
  }

  // ---- phase 3: 48th largest random score among eligible negatives ----
  const unsigned kn = radix_select(1, K_NEG, ovR, ioR, rowbase, posc, rmax,
                                   posT, s_ov, s_io, s_hist, s_bc);

  if (tid == 0) {
    thrA[r] = thrv;
    top2A[r] = t2;
    rmaxA[r] = rmax;
    poscA[r] = posc;
    kthnA[r] = kn;
  }
}

// ---------------- kernel 2: elementwise labeling (float4, L2-fed) ----------------
__global__ __launch_bounds__(TPB) void label_kernel(
    const float4* __restrict__ ov4, const float4* __restrict__ io4,
    const float4* __restrict__ nm4, const float* __restrict__ thrA,
    const float* __restrict__ top2A, const float* __restrict__ rmaxA,
    const unsigned* __restrict__ poscA, const unsigned* __restrict__ kthnA,
    float4* __restrict__ out4, int n4) {
  const int i = blockIdx.x * TPB + threadIdx.x;
  if (i >= n4) return;
  const float4 o = ov4[i];
  const float4 x = io4[i];
  const float4 m = nm4[i];
  const unsigned base = (unsigned)i * 4u;      // flat element index (row-aligned)
  const int r = (int)(base >> 16);             // 65536 elements per row
  const int h = r % 3;
  const float posT = (h == 0) ? 0.6f : ((h == 1) ? 0.7f : 0.8f);
  const float thr = thrA[r];
  const float t2 = top2A[r];
  const float rmax = rmaxA[r];
  const unsigned pc = poscA[r];
  const unsigned kn = kthnA[r];

  const float ovv[4] = {o.x, o.y, o.z, o.w};
  const float iov[4] = {x.x, x.y, x.z, x.w};
  const float nmv[4] = {m.x, m.y, m.z, m.w};
  float res[4];
#pragma unroll
  for (int j = 0; j < 4; ++j) {
    const float v = (pc > 0u) ? ((ovv[j] >= posT) ? iov[j] : 0.0f)
                              : ((iov[j] >= rmax) ? iov[j] : 0.0f);
    const bool pos = (v >= thr) || (v > t2);
    const float posf = pos ? nmv[j] : 0.0f;    // pos * neg_mask
    const bool elig = (ovv[j] <= NEG_T);
    const bool neg = elig && (rand_u32(base + (unsigned)j) >= kn);
    res[j] = -1.0f + (neg ? 1.0f : 0.0f) + 2.0f * posf;
  }
  float4 out;
  out.x = res[0]; out.y = res[1]; out.z = res[2]; out.w = res[3];
  out4[i] = out;
}

// ---------------- host entry ----------------
extern "C" void kernel_launch(void* const* d_in, const int* in_sizes, int n_in,
                              void* d_out, int out_size, void* d_ws,
                              size_t ws_size, hipStream_t stream) {
  (void)in_sizes; (void)n_in; (void)out_size; (void)ws_size;
  const float* ov = (const float*)d_in[0];
  const float* io = (const float*)d_in[1];
  const float* nm = (const float*)d_in[2];
  float* out = (float*)d_out;

  float* wsf = (float*)d_ws;                   // 192*5*4B = 3840B of scratch
  float* thrA = wsf + 0 * ROWS;
  float* top2A = wsf + 1 * ROWS;
  float* rmaxA = wsf + 2 * ROWS;
  unsigned* poscA = (unsigned*)(wsf + 3 * ROWS);
  unsigned* kthnA = (unsigned*)(wsf + 4 * ROWS);

  select_kernel<<<dim3(ROWS), dim3(TPB), 0, stream>>>(ov, io, thrA, top2A,
                                                      rmaxA, poscA, kthnA);

  const int n4 = NELEM / 4;
  label_kernel<<<dim3((n4 + TPB - 1) / TPB), dim3(TPB), 0, stream>>>(
      (const float4*)ov, (const float4*)io, (const float4*)nm, thrA, top2A,
      rmaxA, poscA, kthnA, (float4*)out, n4);
}